// FPS_9612136808568
// MI455X (gfx1250) — compile-verified
//
#include <hip/hip_runtime.h>
#include <stdint.h>

// ---------------------------------------------------------------------------
// FPS (farthest point sampling) + gather for MI455X (gfx1250).
//
// Design (latency-bound sequential scan, NOT a matmul -> no WMMA applies):
//  * 1 workgroup (1024 thr = 32 waves) per batch, whole pos[b] (192KB) parked
//    in the 320KB WGP LDS via the Tensor Data Mover (tensor_load_to_lds,
//    TENSORcnt-tracked), coords + running min-dist held in VGPRs.
//  * per iteration: 16 pts/thread dist update (fma chain), u64 packed
//    (dist_bits<<32 | ~idx) argmax, wave32 xor-butterfly + one workgroup
//    barrier (double-buffered LDS slab for the 32 wave keys).
//  * gather kernel: float4 row copies, bandwidth-trivial (~34MB @ 23.3TB/s).
// ---------------------------------------------------------------------------

#define BATCH 8
#define NPTS  16384
#define NCH   256
#define MOUT  (NPTS / 4)        // 4096
#define KPT   16                // points per thread
#define NTHR  1024              // 32 waves
#define POS_BYTES (NPTS * 3 * 4)  // 192 KB
#define NE    (NPTS * 3)          // 49152 f32 elements (fits 16-bit tile_dim0)

typedef unsigned int uint32x4 __attribute__((ext_vector_type(4)));
typedef int          int32x8  __attribute__((ext_vector_type(8)));
typedef int          int32x4  __attribute__((ext_vector_type(4)));

__device__ __forceinline__ unsigned long long umax64(unsigned long long a,
                                                     unsigned long long b) {
  return a > b ? a : b;
}

__global__ __launch_bounds__(NTHR, 1)
void fps_kernel(const float* __restrict__ pos, int* __restrict__ idxOut) {
  extern __shared__ char smem[];
  float* posL = (float*)smem;                                   // 192 KB
  unsigned long long* red = (unsigned long long*)(smem + POS_BYTES); // 2x32 u64

  const int b    = blockIdx.x;
  const int tid  = (int)threadIdx.x;
  const int lane = tid & 31;
  const int wv   = tid >> 5;
  const float* gpos = pos + (size_t)b * NPTS * 3;

  // ---- stage pos[b] global -> LDS ---------------------------------------
#if __has_builtin(__builtin_amdgcn_tensor_load_to_lds)
  if (tid < 32) {  // wave 0 issues the TDM descriptor once
    unsigned long long ga = (unsigned long long)(uintptr_t)gpos;
    unsigned ldsOff = __builtin_amdgcn_groupstaticsize();  // dynamic LDS base
    uint32x4 g0;
    g0.x = 1u;                                           // count=1, user mode
    g0.y = ldsOff;                                       // lds_addr (bytes)
    g0.z = (unsigned)ga;                                 // global_addr[31:0]
    g0.w = ((unsigned)(ga >> 32) & 0x01FFFFFFu) | 0x80000000u; // type=2
    int32x8 g1;
    g1[0] = (int)0x00020000u;                 // data_size=2 -> 4B, no mcast
    g1[1] = (int)((NE & 0xFFFFu) << 16);      // tensor_dim0[15:0]
    g1[2] = (int)(((NE >> 16) & 0xFFFFu) | (1u << 16)); // dim0 hi | dim1=1
    g1[3] = (int)((NE & 0xFFFFu) << 16);      // tile_dim0 = 49152
    g1[4] = 1;                                // tile_dim1=1, tile_dim2=0
    g1[5] = (int)NE;                          // tensor_dim0_stride lo
    g1[6] = (int)((NE & 0xFFFFu) << 16);      // dim1_stride[15:0]
    g1[7] = (int)(NE >> 16);                  // dim1_stride[47:16]
    int32x4 gz4 = {0, 0, 0, 0};               // <=2D: groups 2/3 unused
    int32x8 gz8 = {0, 0, 0, 0, 0, 0, 0, 0};
    __builtin_amdgcn_tensor_load_to_lds(g0, g1, gz4, gz4, gz8, 0);
    __builtin_amdgcn_s_wait_tensorcnt(0);
  }
#else
  {  // fallback: cooperative vectorized copy
    const float4* g4 = (const float4*)gpos;
    float4* l4 = (float4*)posL;
    for (int i = tid; i < NE / 4; i += NTHR) l4[i] = g4[i];
  }
#endif
  __syncthreads();

  // ---- hot state into registers -----------------------------------------
  float X[KPT], Y[KPT], Z[KPT], D[KPT];
#pragma unroll
  for (int k = 0; k < KPT; ++k) {
    int j = tid + k * NTHR;
    X[k] = posL[3 * j + 0];
    Y[k] = posL[3 * j + 1];
    Z[k] = posL[3 * j + 2];
    D[k] = 1e10f;
  }
  if (tid == 0) idxOut[(size_t)b * MOUT] = 0;   // first index is always 0

  unsigned last = 0;
  int buf = 0;
  for (int it = 1; it < MOUT; ++it) {
    // broadcast read of the last selected point (conflict-free LDS bcast)
    float px = posL[3 * last + 0];
    float py = posL[3 * last + 1];
    float pz = posL[3 * last + 2];

    float best = -1.0f;
    unsigned bidx = 0;
#pragma unroll
    for (int k = 0; k < KPT; ++k) {
      float dx = X[k] - px, dy = Y[k] - py, dz = Z[k] - pz;
      float d = dx * dx;
      d = fmaf(dy, dy, d);
      d = fmaf(dz, dz, d);
      float nd = fminf(D[k], d);
      D[k] = nd;
      if (nd > best) { best = nd; bidx = (unsigned)(tid + k * NTHR); }
    }

    // pack (dist,idx): dist>=0 so float bits are order-preserving; ~idx makes
    // ties pick the SMALLEST index (matches jnp.argmax first-occurrence).
    unsigned long long key =
        ((unsigned long long)__float_as_uint(best) << 32) |
        (unsigned long long)(unsigned)(~bidx);

    // wave32 xor-butterfly reduce
#pragma unroll
    for (int off = 16; off > 0; off >>= 1)
      key = umax64(key, (unsigned long long)__shfl_xor(key, off, 32));

    if (lane == 0) red[buf * 32 + wv] = key;
    __syncthreads();  // single barrier per iteration (double-buffered slab)

    unsigned long long w = red[buf * 32 + lane];
#pragma unroll
    for (int off = 16; off > 0; off >>= 1)
      w = umax64(w, (unsigned long long)__shfl_xor(w, off, 32));

    unsigned nxt = ~(unsigned)w;
    if (tid == 0) idxOut[(size_t)b * MOUT + it] = (int)nxt;
    last = nxt;
    buf ^= 1;
  }
}

// ---- gather: one 64-thread block per output row -------------------------
__global__ __launch_bounds__(64)
void gather_kernel(const float* __restrict__ pos, const float* __restrict__ feat,
                   const int* __restrict__ idx, float* __restrict__ posOut,
                   float* __restrict__ featOut) {
  int row = (int)blockIdx.x;          // b*MOUT + i
  int b   = row >> 12;                // MOUT = 4096
  int t   = (int)threadIdx.x;
  int j   = idx[row];

  const float4* src = (const float4*)(feat + ((size_t)b * NPTS + j) * NCH);
  float4* dst = (float4*)(featOut + (size_t)row * NCH);
  __builtin_prefetch(src + t, 0, 0);  // global_prefetch_b8
  dst[t] = src[t];                    // 64 lanes x float4 = 256 ch

  if (t < 3)
    posOut[(size_t)row * 3 + t] = pos[((size_t)b * NPTS + j) * 3 + t];
}

extern "C" void kernel_launch(void* const* d_in, const int* in_sizes, int n_in,
                              void* d_out, int out_size, void* d_ws,
                              size_t ws_size, hipStream_t stream) {
  (void)in_sizes; (void)n_in; (void)out_size; (void)ws_size;
  const float* pos  = (const float*)d_in[0];   // [B,N,3] f32
  const float* feat = (const float*)d_in[1];   // [B,N,C] f32
  int* idx = (int*)d_ws;                       // [B,M] i32 scratch (128 KB)

  float* out     = (float*)d_out;
  float* posOut  = out;                        // [B,M,3]
  float* featOut = out + (size_t)BATCH * MOUT * 3;  // [B,M,C]

  // 192KB pos slab + 512B reduction slab of dynamic LDS (WGP has 320KB)
  (void)hipFuncSetAttribute((const void*)fps_kernel,
                            hipFuncAttributeMaxDynamicSharedMemorySize,
                            POS_BYTES + 512);
  fps_kernel<<<BATCH, NTHR, POS_BYTES + 512, stream>>>(pos, idx);
  gather_kernel<<<BATCH * MOUT, 64, 0, stream>>>(pos, feat, idx, posOut,
                                                 featOut);
}